// TripletLoss_10325101379760
// MI455X (gfx1250) — compile-verified
//
#include <hip/hip_runtime.h>

typedef float v2f __attribute__((ext_vector_type(2)));
typedef float v8f __attribute__((ext_vector_type(8)));

#define N 512
#define D 512
#define EPSF 1e-8f
#define MARGINF 1.0f

// ---------------- Kernel 1: row L2-normalize (norm clamped at eps) ----------
__global__ void __launch_bounds__(256) normalize_rows(const float* __restrict__ embs,
                                                      float* __restrict__ en) {
    __shared__ float red[256];
    __shared__ float rn;
    const int row = blockIdx.x;
    const int tid = threadIdx.x;
    const float* r = embs + (size_t)row * D;

    float s = 0.f;
    for (int c = tid; c < D; c += 256) {
        float v = r[c];
        s += v * v;
    }
    red[tid] = s;
    __syncthreads();
    for (int off = 128; off > 0; off >>= 1) {
        if (tid < off) red[tid] += red[tid + off];
        __syncthreads();
    }
    if (tid == 0) {
        float nrm = fmaxf(sqrtf(red[0]), EPSF);
        rn = 1.0f / nrm;
    }
    __syncthreads();
    const float rr = rn;
    for (int c = tid; c < D; c += 256) {
        en[(size_t)row * D + c] = r[c] * rr;
    }
}

// ---------------- Kernel 2: cos = en @ en^T via V_WMMA_F32_16X16X4_F32 ------
// One wave32 per 16x64 output strip: one A fragment feeds 4 B fragments /
// 4 accumulators per K-step (5 loads per 4 WMMA instead of 8). Strips
// strictly below the diagonal are skipped: the consumer only reads j > i.
// A fragment (16x4):  lanes 0-15 hold K = k0+0/1, lanes 16-31 hold K = k0+2/3.
// B fragment (4x16):  mirrored layout; B[k,n] = en[n, k] since B = en^T.
// C/D 16x16 f32: VGPR r holds row (r + 8*half) at column = lane%16.
__global__ void __launch_bounds__(32) cos_gemm_wmma(const float* __restrict__ en,
                                                    float* __restrict__ cosm) {
    const int m0 = blockIdx.y * 16;
    const int n0 = blockIdx.x * 64;
    if (n0 + 63 < m0) return;          // whole strip below diagonal: never read

    const int lane = threadIdx.x;
    const int half = lane >> 4;        // 0 or 1
    const int l    = lane & 15;
    const int koff = half * 2;         // K sub-offset per half-wave

    const float* arow = en + (size_t)(m0 + l) * D + koff;
    const float* b0   = en + (size_t)(n0 +  0 + l) * D + koff;
    const float* b1   = en + (size_t)(n0 + 16 + l) * D + koff;
    const float* b2   = en + (size_t)(n0 + 32 + l) * D + koff;
    const float* b3   = en + (size_t)(n0 + 48 + l) * D + koff;

    v8f c0 = {}, c1 = {}, c2 = {}, c3 = {};
#pragma unroll 2
    for (int k0 = 0; k0 < D; k0 += 4) {
        const v2f a  = *(const v2f*)(arow + k0);
        const v2f q0 = *(const v2f*)(b0 + k0);
        const v2f q1 = *(const v2f*)(b1 + k0);
        const v2f q2 = *(const v2f*)(b2 + k0);
        const v2f q3 = *(const v2f*)(b3 + k0);
        // 8 args: (neg_a, A, neg_b, B, c_mod, C, reuse_a, reuse_b)
        c0 = __builtin_amdgcn_wmma_f32_16x16x4_f32(false, a, false, q0, (short)0, c0, false, false);
        c1 = __builtin_amdgcn_wmma_f32_16x16x4_f32(false, a, false, q1, (short)0, c1, false, false);
        c2 = __builtin_amdgcn_wmma_f32_16x16x4_f32(false, a, false, q2, (short)0, c2, false, false);
        c3 = __builtin_amdgcn_wmma_f32_16x16x4_f32(false, a, false, q3, (short)0, c3, false, false);
    }

    float* dst = cosm + (size_t)(m0 + half * 8) * N + n0 + l;
#pragma unroll
    for (int r = 0; r < 8; ++r) {
        dst[(size_t)r * N +  0] = c0[r];
        dst[(size_t)r * N + 16] = c1[r];
        dst[(size_t)r * N + 32] = c2[r];
        dst[(size_t)r * N + 48] = c3[r];
    }
}

// ---------------- Kernel 3 prep: zero the scalar output ---------------------
__global__ void zero_out(float* __restrict__ out) {
    if (threadIdx.x == 0 && blockIdx.x == 0) out[0] = 0.f;
}

// ---------------- Kernel 3: masked triplet relu-sum -------------------------
// Block per anchor i. Compact cos[i, j>i] into LDS positive / negative lists
// by class, then sum relu(neg[k] - pos[j] + margin) over the pair product.
// P is tiny (samples per class / 2 on average), so j is the outer loop and
// the negative list is swept thread-strided out of LDS: no divisions.
__global__ void __launch_bounds__(256) triplet_sum(const float* __restrict__ cosm,
                                                   const int* __restrict__ idx,
                                                   float* __restrict__ out) {
    __shared__ float posv[N];
    __shared__ float negv[N];
    __shared__ int   pcnt, ncnt;
    __shared__ float red[256];

    const int i   = blockIdx.x;
    const int tid = threadIdx.x;
    if (tid == 0) { pcnt = 0; ncnt = 0; }
    __syncthreads();

    const int my = idx[i];
    for (int j = i + 1 + tid; j < N; j += 256) {
        const float v = cosm[(size_t)i * N + j];
        if (idx[j] == my) {
            const int p = atomicAdd(&pcnt, 1);
            posv[p] = v;
        } else {
            const int p = atomicAdd(&ncnt, 1);
            negv[p] = v;
        }
    }
    __syncthreads();

    const int P  = pcnt;
    const int Ng = ncnt;
    float s = 0.f;
    for (int j = 0; j < P; ++j) {
        const float t = MARGINF - posv[j];
        for (int k = tid; k < Ng; k += 256) {
            s += fmaxf(negv[k] + t, 0.f);
        }
    }

    red[tid] = s;
    __syncthreads();
    for (int off = 128; off > 0; off >>= 1) {
        if (tid < off) red[tid] += red[tid + off];
        __syncthreads();
    }
    if (tid == 0) atomicAdd(out, red[0]);
}

// ---------------- Host-side launcher ----------------------------------------
extern "C" void kernel_launch(void* const* d_in, const int* in_sizes, int n_in,
                              void* d_out, int out_size, void* d_ws, size_t ws_size,
                              hipStream_t stream) {
    const float* embs = (const float*)d_in[0];
    const int*   idx  = (const int*)d_in[1];
    float*       out  = (float*)d_out;

    float* en   = (float*)d_ws;        // 512*512 f32 = 1 MB
    float* cosm = en + (size_t)N * D;  // 512*512 f32 = 1 MB

    normalize_rows<<<N, 256, 0, stream>>>(embs, en);

    dim3 grid(N / 64, N / 16);         // 16x64 strip per wave32
    cos_gemm_wmma<<<grid, 32, 0, stream>>>(en, cosm);

    zero_out<<<1, 64, 0, stream>>>(out);
    triplet_sum<<<N, 256, 0, stream>>>(cosm, idx, out);
}